// HeightVoxelLoss_27934467293391
// MI455X (gfx1250) — compile-verified
//
#include <hip/hip_runtime.h>
#include <hip/hip_bf16.h>

// HeightVoxelLoss for MI455X (gfx1250, wave32).
// Gather-bound scalar-loss reduction: ~9 MB of gathered reads, trivial FLOPs.
// Deterministic: integer atomics only; float sums use fixed-shape trees.
// Wave-level reduction runs on the matrix unit via V_WMMA_F32_16X16X4_F32
// (the only full-f32-precision WMMA shape), B = all-ones.

typedef __attribute__((ext_vector_type(2))) float v2f;
typedef __attribute__((ext_vector_type(8))) float v8f;

#define BB     2
#define XX     200
#define YY     200
#define ZZ     16
#define CC     17
#define NSEL   4000
#define EMPTY  16

// Sum x over all 32 lanes using the matrix unit.
// A-matrix (16x4 f32): lane L(0..15): vgpr0=A[L][0], vgpr1=A[L][1];
//                      lane 16+L:     vgpr0=A[L][2], vgpr1=A[L][3].
// With a = {x, 0} and B = ones: D[m][*] = x_m + x_{16+m}.
// D layout: lanes 0..15 hold rows 0..7 in their 8 VGPRs, lanes 16..31 rows 8..15.
// Per-lane sum of the 8 VGPRs gives a half-wave sum; shfl_xor(16) completes it.
__device__ __forceinline__ float wave_sum_wmma(float x) {
  v2f a; a[0] = x;   a[1] = 0.0f;
  v2f b; b[0] = 1.0f; b[1] = 1.0f;
  v8f c = {};
  v8f d = __builtin_amdgcn_wmma_f32_16x16x4_f32(
      /*neg_a=*/false, a, /*neg_b=*/false, b,
      /*c_mod=*/(short)0, c, /*reuse_a=*/false, /*reuse_b=*/false);
  float s = d[0] + d[1] + d[2] + d[3] + d[4] + d[5] + d[6] + d[7];
  s += __shfl_xor(s, 16, 32);
  return s;  // all lanes hold the wave total
}

__global__ void hv_zero_counts(int* __restrict__ counts) {
  if (threadIdx.x < BB * ZZ) counts[threadIdx.x] = 0;
}

// counts[b][z] = #selected voxels at height z with label != EMPTY
__global__ void hv_count(const int* __restrict__ labels,
                         const int* __restrict__ sel,
                         int* __restrict__ counts) {
  __shared__ int scnt[ZZ];
  const int b   = blockIdx.y;
  const int tid = threadIdx.x;
  if (tid < ZZ) scnt[tid] = 0;
  __syncthreads();
  const int col = blockIdx.x * blockDim.x + tid;
  if (col < NSEL) {
    const int xs = sel[(b * NSEL + col) * 2 + 0];
    const int ys = sel[(b * NSEL + col) * 2 + 1];
    const int* lp = labels + ((size_t)((b * XX + xs) * YY + ys)) * ZZ;
#pragma unroll
    for (int z = 0; z < ZZ; ++z) {
      if (lp[z] != EMPTY) atomicAdd(&scnt[z], 1);
    }
  }
  __syncthreads();
  if (tid < ZZ) atomicAdd(&counts[b * ZZ + tid], scnt[tid]);
}

// One thread per voxel (b, col, z). 250 blocks x 256 threads per batch, exact.
__global__ void hv_loss(const float* __restrict__ preds,
                        const int* __restrict__ labels,
                        const int* __restrict__ sel,
                        const int* __restrict__ counts,
                        float* __restrict__ partials) {
  const int b     = blockIdx.y;
  const int tid   = threadIdx.x;
  const int voxel = blockIdx.x * blockDim.x + tid;  // 0..63999
  const int col   = voxel >> 4;
  const int z     = voxel & 15;

  // weights[b][z] = counts>0 ? MAX_W * (MIN_W/MAX_W)^(c/max(max_c,1)) : 0
  int cmax = 1;
#pragma unroll
  for (int k = 0; k < ZZ; ++k) cmax = max(cmax, counts[b * ZZ + k]);
  const int   cz = counts[b * ZZ + z];
  const float w  = (cz > 0)
      ? 3.0f * expf(-1.0986122886681098f * ((float)cz / (float)cmax))
      : 0.0f;

  const int xs = sel[(b * NSEL + col) * 2 + 0];
  const int ys = sel[(b * NSEL + col) * 2 + 1];
  const size_t vox = ((size_t)((b * XX + xs) * YY + ys)) * ZZ + z;
  const int lab = labels[vox];
  const float* lp = preds + vox * CC;  // 68-byte contiguous, 4B aligned

  float l[CC];
#pragma unroll
  for (int c2 = 0; c2 < CC; ++c2) l[c2] = lp[c2];
  float m = l[0];
#pragma unroll
  for (int c2 = 1; c2 < CC; ++c2) m = fmaxf(m, l[c2]);
  float s = 0.0f;
#pragma unroll
  for (int c2 = 0; c2 < CC; ++c2) s += expf(l[c2] - m);

  float contrib = 0.0f;
  if (lab != EMPTY) {
    const float p  = expf(l[lab] - m) / s;
    const float wl = w * logf(p + 0.001f);
    const float aw = fabsf(wl);
    contrib = (aw < 1.0f) ? 0.5f * wl * wl : aw - 0.5f;
  }

  // Block reduce: WMMA wave sum -> LDS -> thread 0 (fixed order, deterministic).
  const float wavesum = wave_sum_wmma(contrib);
  __shared__ float sh[8];
  const int lane = tid & 31, wave = tid >> 5;
  if (lane == 0) sh[wave] = wavesum;
  __syncthreads();
  if (tid == 0) {
    float t = 0.0f;
#pragma unroll
    for (int wv = 0; wv < 8; ++wv) t += sh[wv];
    partials[b * gridDim.x + blockIdx.x] = t;
  }
}

// Single wave: deterministic fold of per-block partials, normalize, mean over B.
__global__ void hv_finalize(const int* __restrict__ counts,
                            const float* __restrict__ partials,
                            int nblocks,
                            float* __restrict__ out) {
  const int lane = threadIdx.x;  // 0..31
  float loss = 0.0f;
  for (int b = 0; b < BB; ++b) {
    float s = 0.0f;
    for (int i = lane; i < nblocks; i += 32) s += partials[b * nblocks + i];
    const float tot = wave_sum_wmma(s);
    int nv = 0;
#pragma unroll
    for (int zz = 0; zz < ZZ; ++zz) nv += counts[b * ZZ + zz];
    loss += tot / fmaxf((float)nv, 1.0f);
  }
  if (lane == 0) out[0] = loss * (1.0f / (float)BB);
}

extern "C" void kernel_launch(void* const* d_in, const int* in_sizes, int n_in,
                              void* d_out, int out_size, void* d_ws, size_t ws_size,
                              hipStream_t stream) {
  (void)in_sizes; (void)n_in; (void)out_size; (void)ws_size;
  const float* preds  = (const float*)d_in[0];
  const int*   labels = (const int*)d_in[1];
  const int*   sel    = (const int*)d_in[2];

  int*   counts   = (int*)d_ws;                         // 32 ints
  float* partials = (float*)((char*)d_ws + 128);        // B * 250 floats
  float* out      = (float*)d_out;

  hv_zero_counts<<<1, 64, 0, stream>>>(counts);
  hv_count<<<dim3((NSEL + 255) / 256, BB), 256, 0, stream>>>(labels, sel, counts);

  const int nblocks = (NSEL * ZZ) / 256;  // 250, exact
  hv_loss<<<dim3(nblocks, BB), 256, 0, stream>>>(preds, labels, sel, counts, partials);
  hv_finalize<<<1, 32, 0, stream>>>(counts, partials, nblocks, out);
}